// EncoderDecoder_10866267259014
// MI455X (gfx1250) — compile-verified
//
#include <hip/hip_runtime.h>
#include <math.h>

// Problem constants (from reference)
#define NODE_V 16
#define EDGE_V 8
#define B_DIM  64
#define N_DIM  256

// Per-batch edge elements: N*N = 65536. 64 edge-blocks/batch * 256 thr * 4 elem = 65536.
#define EDGE_BLOCKS_PER_BATCH 64

typedef float v4f __attribute__((ext_vector_type(4)));

__global__ __launch_bounds__(256) void enc_dec_logprob_kernel(
    const float* __restrict__ x_nodes,   // [B, N, 1]
    const float* __restrict__ x_edges,   // [B, N, N]
    const float* __restrict__ g_0,       // [B]
    float* __restrict__ out_nodes,       // [B, N, NODE_V]
    float* __restrict__ out_edges)       // [B, N, N, EDGE_V]
{
    // LUTs in LDS: row-per-bucket log-softmax'd logits.
    __shared__ __align__(16) float lutE[9 * EDGE_V];    // 9 rows x 8  (32 B rows)
    __shared__ __align__(16) float lutN[17 * NODE_V];   // 17 rows x 16 (64 B rows)

    const int b       = blockIdx.y;
    const int tid     = threadIdx.x;
    const bool isEdge = (blockIdx.x < EDGE_BLOCKS_PER_BATCH);

    // ---- Issue streaming input load FIRST so its VMEM latency hides behind
    //      the LUT build below (compiler won't hoist loads across the barrier).
    //      Non-temporal: read-once data, don't pollute WGP$/L2.
    size_t base = 0;
    v4f x4 = {0.f, 0.f, 0.f, 0.f};
    float xnv = 0.f;
    if (isEdge) {
        base = ((size_t)b << 16) + (size_t)(blockIdx.x * 1024 + tid * 4);
        x4 = __builtin_nontemporal_load(
                 reinterpret_cast<const v4f*>(x_edges + base));
    } else {
        xnv = x_nodes[(size_t)b * N_DIM + (size_t)tid];
    }

    // inv_stdev = exp(-0.5 * g0[b])  (uniform per block -> scalar path)
    const float inv = expf(-0.5f * g_0[b]);

    // ---- Build LUTs (one thread per bucket row; cold, uniform-ish math) ----
    if (tid < 9) {
        // edge row k = tid: diff = (2/EDGE_V)*inv*(k - v)
        const float s = inv * (2.0f / (float)EDGE_V);
        float l[EDGE_V];
        float m = -3.0e38f;
#pragma unroll
        for (int v = 0; v < EDGE_V; ++v) {
            const float d = s * (float)(tid - v);
            l[v] = -0.5f * d * d;
            m = fmaxf(m, l[v]);
        }
        float sum = 0.0f;
#pragma unroll
        for (int v = 0; v < EDGE_V; ++v) sum += expf(l[v] - m);
        const float lse = m + logf(sum);
#pragma unroll
        for (int v = 0; v < EDGE_V; ++v) lutE[tid * EDGE_V + v] = l[v] - lse;
    } else if (tid >= 32 && tid < 32 + 17) {
        // node row k = tid-32: diff = (2/NODE_V)*inv*(k - v)
        const int k = tid - 32;
        const float s = inv * (2.0f / (float)NODE_V);
        float l[NODE_V];
        float m = -3.0e38f;
#pragma unroll
        for (int v = 0; v < NODE_V; ++v) {
            const float d = s * (float)(k - v);
            l[v] = -0.5f * d * d;
            m = fmaxf(m, l[v]);
        }
        float sum = 0.0f;
#pragma unroll
        for (int v = 0; v < NODE_V; ++v) sum += expf(l[v] - m);
        const float lse = m + logf(sum);
#pragma unroll
        for (int v = 0; v < NODE_V; ++v) lutN[k * NODE_V + v] = l[v] - lse;
    }
    __syncthreads();

    if (isEdge) {
        // ---- Edge streaming: 4 consecutive elements/thread.
        // Per element: v_rndne+cvt, 2x ds_load_b128 (same-address lanes broadcast),
        // 2x global_store_b128 NT. 128 B contiguous output per thread.
        const int k0 = (int)rintf(x4.x);
        const int k1 = (int)rintf(x4.y);
        const int k2 = (int)rintf(x4.z);
        const int k3 = (int)rintf(x4.w);

        const v4f* le = reinterpret_cast<const v4f*>(lutE);  // 2 v4f per row
        v4f* o = reinterpret_cast<v4f*>(out_edges + base * EDGE_V);

        __builtin_nontemporal_store(le[k0 * 2 + 0], o + 0);
        __builtin_nontemporal_store(le[k0 * 2 + 1], o + 1);
        __builtin_nontemporal_store(le[k1 * 2 + 0], o + 2);
        __builtin_nontemporal_store(le[k1 * 2 + 1], o + 3);
        __builtin_nontemporal_store(le[k2 * 2 + 0], o + 4);
        __builtin_nontemporal_store(le[k2 * 2 + 1], o + 5);
        __builtin_nontemporal_store(le[k3 * 2 + 0], o + 6);
        __builtin_nontemporal_store(le[k3 * 2 + 1], o + 7);
    } else {
        // ---- Node streaming: one element per thread (N_DIM == blockDim.x) ----
        const size_t idx = (size_t)b * N_DIM + (size_t)tid;
        const int k = (int)rintf(xnv);

        const v4f* ln = reinterpret_cast<const v4f*>(lutN);  // 4 v4f per row
        v4f* o = reinterpret_cast<v4f*>(out_nodes + idx * NODE_V);
        __builtin_nontemporal_store(ln[k * 4 + 0], o + 0);
        __builtin_nontemporal_store(ln[k * 4 + 1], o + 1);
        __builtin_nontemporal_store(ln[k * 4 + 2], o + 2);
        __builtin_nontemporal_store(ln[k * 4 + 3], o + 3);
    }
}

extern "C" void kernel_launch(void* const* d_in, const int* in_sizes, int n_in,
                              void* d_out, int out_size, void* d_ws, size_t ws_size,
                              hipStream_t stream) {
    (void)in_sizes; (void)n_in; (void)out_size; (void)d_ws; (void)ws_size;

    const float* x_nodes = (const float*)d_in[0];  // [64,256,1]
    const float* x_edges = (const float*)d_in[1];  // [64,256,256]
    const float* g0      = (const float*)d_in[2];  // [64]

    float* out_nodes = (float*)d_out;                                  // [64,256,16]
    float* out_edges = (float*)d_out + (size_t)B_DIM * N_DIM * NODE_V; // [64,256,256,8]

    dim3 grid(EDGE_BLOCKS_PER_BATCH + 1, B_DIM);  // 64 edge blocks + 1 node block, per batch
    dim3 block(256);
    enc_dec_logprob_kernel<<<grid, block, 0, stream>>>(x_nodes, x_edges, g0,
                                                       out_nodes, out_edges);
}